// GABlock_83786222011204
// MI455X (gfx1250) — compile-verified
//
#include <hip/hip_runtime.h>
#include <hip/hip_bf16.h>
#include <math.h>

// ---- problem constants (match reference) ----
constexpr int NN  = 20000;     // nodes
constexpr int NE  = 320000;    // edges
constexpr int DD  = 128;       // node dim
constexpr int CC  = 32;        // edge dim
constexpr int HH  = 12;        // heads
constexpr int QKD = 32;        // qk dim
constexpr int VD  = 32;        // value dim
constexpr int QO  = HH * QKD;      // 384   q/k/v width
constexpr int PO  = HH * 8 * 3;    // 288   point proj width
constexpr int FIN = HH*CC + HH*VD + HH*8*7; // 1440

typedef float v2f __attribute__((ext_vector_type(2)));
typedef float v8f __attribute__((ext_vector_type(8)));

// ---------------------------------------------------------------------------
// fp32 WMMA GEMM, 2-wide N register blocking: each wave computes a 16x32
// output tile (two V_WMMA_F32_16X16X4_F32 accumulators sharing the A frag).
// Requires M%16==0, Nc%32==0, K%4==0 (true for all GEMMs here).
// epi: 0=none, 1=+bias, 2=+bias,relu
// ---------------------------------------------------------------------------
__global__ void k_gemm_wmma(const float* __restrict__ A, const float* __restrict__ B,
                            float* __restrict__ C, const float* __restrict__ bias,
                            int M, int K, int Nc, int epi) {
    const int wid    = threadIdx.x >> 5;
    const int lane   = threadIdx.x & 31;
    const int tilesN = Nc >> 5;                       // 32-wide super-tiles
    const int total  = (M >> 4) * tilesN;
    const int tile   = blockIdx.x * (blockDim.x >> 5) + wid;
    if (tile >= total) return;                        // wave-uniform exit
    const int tm = tile / tilesN;
    const int tn = tile - tm * tilesN;

    const int row   = (tm << 4) + (lane & 15);        // A row held by this lane
    const int col0  = (tn << 5) + (lane & 15);        // first 16-col tile
    const int col1  = col0 + 16;                      // second 16-col tile
    const int khalf = (lane >> 4) << 1;               // 0 (lanes 0-15) or 2 (16-31)

    const float* __restrict__ arow = A + (size_t)row * K;
    v8f acc0 = {}, acc1 = {};
    for (int k = 0; k < K; k += 4) {
        const int kk = k + khalf;
        v2f a = *(const v2f*)(arow + kk);             // A[row][kk], A[row][kk+1]
        const float* __restrict__ br0 = B + (size_t)kk       * Nc;
        const float* __restrict__ br1 = B + (size_t)(kk + 1) * Nc;
        v2f b0; b0.x = br0[col0]; b0.y = br1[col0];
        v2f b1; b1.x = br0[col1]; b1.y = br1[col1];
        acc0 = __builtin_amdgcn_wmma_f32_16x16x4_f32(
                   false, a, false, b0, (short)0, acc0, false, false);
        acc1 = __builtin_amdgcn_wmma_f32_16x16x4_f32(
                   false, a, false, b1, (short)0, acc1, false, false);
    }
    const float bv0 = (epi >= 1) ? bias[col0] : 0.0f;
    const float bv1 = (epi >= 1) ? bias[col1] : 0.0f;
    const int rbase = (tm << 4) + ((lane >> 4) << 3); // M = vgpr + 8*lane[4]
    #pragma unroll
    for (int i = 0; i < 8; ++i) {
        float v0 = acc0[i] + bv0;
        float v1 = acc1[i] + bv1;
        if (epi == 2) { v0 = fmaxf(v0, 0.0f); v1 = fmaxf(v1, 0.0f); }
        C[(size_t)(rbase + i) * Nc + col0] = v0;
        C[(size_t)(rbase + i) * Nc + col1] = v1;
    }
}

// ---------------------------------------------------------------------------
__global__ void k_fill(float* __restrict__ p, float v, int n) {
    int i = blockIdx.x * blockDim.x + threadIdx.x;
    if (i < n) p[i] = v;
}

// local->global in place: p <- R p + t  (thread per 3-vector point)
__global__ void k_l2g(float* __restrict__ p, const float* __restrict__ R,
                      const float* __restrict__ t, int npts /* = NN*96 */) {
    int i = blockIdx.x * blockDim.x + threadIdx.x;
    if (i >= npts) return;
    int node = i / 96;
    const float* Rn = R + (size_t)node * 9;
    const float* tn = t + (size_t)node * 3;
    float* pp = p + (size_t)i * 3;
    float x = pp[0], y = pp[1], z = pp[2];
    pp[0] = Rn[0]*x + Rn[1]*y + Rn[2]*z + tn[0];
    pp[1] = Rn[3]*x + Rn[4]*y + Rn[5]*z + tn[1];
    pp[2] = Rn[6]*x + Rn[7]*y + Rn[8]*z + tn[2];
}

__device__ inline void atomicMaxFloat(float* addr, float val) {
    if (val >= 0.0f) atomicMax((int*)addr, __float_as_int(val));
    else             atomicMin((unsigned int*)addr, __float_as_uint(val));
}

// logits per (edge, head); also scatter-max into mbuf[src,h]
__global__ void k_logits(const float* __restrict__ q, const float* __restrict__ k,
                         const float* __restrict__ qp, const float* __restrict__ kp,
                         const float* __restrict__ z, const float* __restrict__ Wpair,
                         const float* __restrict__ sc, const long long* __restrict__ ei,
                         float* __restrict__ logits, float* __restrict__ mbuf) {
    int idx = blockIdx.x * blockDim.x + threadIdx.x;
    if (idx >= NE * HH) return;
    int e = idx / HH, h = idx - e * HH;
    int src = (int)ei[e];
    int dst = (int)ei[(size_t)NE + e];

    // node logit: q[src,h,:] . k[dst,h,:] / sqrt(32)
    const float4* qr = (const float4*)(q + (size_t)src * QO + h * QKD);
    const float4* kr = (const float4*)(k + (size_t)dst * QO + h * QKD);
    float dn = 0.0f;
    #pragma unroll
    for (int i = 0; i < 8; ++i) {
        float4 a = qr[i], b = kr[i];
        dn += a.x*b.x + a.y*b.y + a.z*b.z + a.w*b.w;
    }
    // pair logit: z[e,:] . Wpair[:,h]
    const float* zr = z + (size_t)e * CC;
    float dp = 0.0f;
    #pragma unroll 4
    for (int c = 0; c < CC; ++c) dp += zr[c] * Wpair[c * HH + h];
    // spatial: sum_sq(qp[src,h,:] - kp[dst,h,:]) over 24 dims
    const float4* qpr = (const float4*)(qp + (size_t)src * PO + h * 24);
    const float4* kpr = (const float4*)(kp + (size_t)dst * PO + h * 24);
    float ss = 0.0f;
    #pragma unroll
    for (int i = 0; i < 6; ++i) {
        float4 a = qpr[i], b = kpr[i];
        float dx = a.x-b.x, dy = a.y-b.y, dz = a.z-b.z, dw = a.w-b.w;
        ss += dx*dx + dy*dy + dz*dz + dw*dw;
    }
    float gamma = log1pf(expf(sc[h]));                 // softplus
    // sqrt(2/(9*QP))/2 = (1/6)/2 = 1/12
    float lg = (dn * 0.17677669529663687f + dp + ss * (-gamma * (1.0f/12.0f)))
               * 0.5773502691896258f;                   // * sqrt(1/3)
    logits[idx] = lg;
    atomicMaxFloat(mbuf + (size_t)src * HH + h, lg);
}

// Unnormalized scatter aggregation: w = exp(logit - max); accumulate w into
// sbuf[src,h] and w-weighted z, v, vp into feat_all/aggr. Normalization by
// sbuf happens later at node granularity (denominator is per-(src,h)).
__global__ void k_aggregate(const float* __restrict__ logits, const float* __restrict__ mbuf,
                            float* __restrict__ sbuf,
                            const float* __restrict__ z, const float* __restrict__ v,
                            const float* __restrict__ vp, const long long* __restrict__ ei,
                            float* __restrict__ feat_all, float* __restrict__ aggr) {
    int idx = blockIdx.x * blockDim.x + threadIdx.x;
    if (idx >= NE * HH) return;
    int e = idx / HH, h = idx - e * HH;
    int src = (int)ei[e];
    int dst = (int)ei[(size_t)NE + e];
    float w = expf(logits[idx] - mbuf[(size_t)src * HH + h]);   // <= 1, stable
    atomicAdd(sbuf + (size_t)src * HH + h, w);

    float* fa = feat_all + (size_t)src * FIN;
    const float* zr = z + (size_t)e * CC;
    #pragma unroll 8
    for (int c = 0; c < CC; ++c) atomicAdd(fa + h * CC + c, w * zr[c]);
    const float* vr = v + (size_t)dst * QO + h * VD;
    #pragma unroll 8
    for (int j = 0; j < VD; ++j) atomicAdd(fa + HH*CC + h * VD + j, w * vr[j]);
    const float* vpr = vp + (size_t)dst * PO + h * 24;
    float* ag = aggr + (size_t)src * PO + h * 24;
    #pragma unroll 8
    for (int j = 0; j < 24; ++j) atomicAdd(ag + j, w * vpr[j]);
}

// divide feat_all[:, 0:768] (p2n + node slices) by the softmax denominator
__global__ void k_alpha_norm(float* __restrict__ feat_all, const float* __restrict__ sbuf) {
    int i = blockIdx.x * blockDim.x + threadIdx.x;      // NN * 768
    if (i >= NN * 768) return;
    int node = i / 768, c = i - node * 768;
    int h = (c >> 5) % HH;            // slices are [h*32] then [384 + h*32]
    feat_all[(size_t)node * FIN + c] /= sbuf[(size_t)node * HH + h];
}

// normalize aggregated points, global->local, norm/dir, write feat_spatial
__global__ void k_g2l_finish(const float* __restrict__ aggr, const float* __restrict__ sbuf,
                             const float* __restrict__ R, const float* __restrict__ t,
                             float* __restrict__ feat_all) {
    int i = blockIdx.x * blockDim.x + threadIdx.x;      // NN*96 points
    if (i >= NN * 96) return;
    int node = i / 96, hp = i - node * 96;
    const float* a  = aggr + (size_t)node * PO + hp * 3;
    const float* Rn = R + (size_t)node * 9;
    const float* tn = t + (size_t)node * 3;
    float inv_s = 1.0f / sbuf[(size_t)node * HH + (hp >> 3)];   // h = hp / 8
    float q0 = a[0] * inv_s - tn[0];
    float q1 = a[1] * inv_s - tn[1];
    float q2 = a[2] * inv_s - tn[2];
    float lx = Rn[0]*q0 + Rn[3]*q1 + Rn[6]*q2;          // R^T (q - t)
    float ly = Rn[1]*q0 + Rn[4]*q1 + Rn[7]*q2;
    float lz = Rn[2]*q0 + Rn[5]*q1 + Rn[8]*q2;
    float nr  = sqrtf(lx*lx + ly*ly + lz*lz);
    float inv = 1.0f / fmaxf(nr, 1e-8f);
    float* fa = feat_all + (size_t)node * FIN;
    fa[768  + hp*3 + 0] = lx;  fa[768  + hp*3 + 1] = ly;  fa[768  + hp*3 + 2] = lz;
    fa[1056 + hp]       = nr;
    fa[1152 + hp*3 + 0] = lx*inv; fa[1152 + hp*3 + 1] = ly*inv; fa[1152 + hp*3 + 2] = lz*inv;
}

// out = LayerNorm(a + b) * g + beta   (one 128-wide row per block)
__global__ void k_ln(const float* __restrict__ a, const float* __restrict__ b,
                     const float* __restrict__ g, const float* __restrict__ beta,
                     float* __restrict__ out) {
    __shared__ float p1[4], p2[4];
    int row = blockIdx.x, d = threadIdx.x;
    float r = a[(size_t)row * DD + d] + b[(size_t)row * DD + d];
    float s1 = r, s2 = r * r;
    #pragma unroll
    for (int off = 16; off > 0; off >>= 1) {
        s1 += __shfl_xor(s1, off);
        s2 += __shfl_xor(s2, off);
    }
    if ((threadIdx.x & 31) == 0) { p1[threadIdx.x >> 5] = s1; p2[threadIdx.x >> 5] = s2; }
    __syncthreads();
    float t1 = p1[0] + p1[1] + p1[2] + p1[3];
    float t2 = p2[0] + p2[1] + p2[2] + p2[3];
    float mu  = t1 * (1.0f / DD);
    float var = t2 * (1.0f / DD) - mu * mu;
    out[(size_t)row * DD + d] = (r - mu) * rsqrtf(var + 1e-5f) * g[d] + beta[d];
}

// ---------------------------------------------------------------------------
static inline int cdiv(int a, int b) { return (a + b - 1) / b; }

extern "C" void kernel_launch(void* const* d_in, const int* in_sizes, int n_in,
                              void* d_out, int out_size, void* d_ws, size_t ws_size,
                              hipStream_t stream) {
    const float* R     = (const float*)d_in[0];
    const float* t     = (const float*)d_in[1];
    const float* x     = (const float*)d_in[2];
    const float* z     = (const float*)d_in[3];
    const long long* ei= (const long long*)d_in[4];
    const float* Wq    = (const float*)d_in[5];
    const float* Wk    = (const float*)d_in[6];
    const float* Wv    = (const float*)d_in[7];
    const float* Wpair = (const float*)d_in[8];
    const float* sc    = (const float*)d_in[9];
    const float* Wqp   = (const float*)d_in[10];
    const float* Wkp   = (const float*)d_in[11];
    const float* Wvp   = (const float*)d_in[12];
    const float* Wo    = (const float*)d_in[13];
    const float* bo    = (const float*)d_in[14];
    const float* ln1g  = (const float*)d_in[15];
    const float* ln1b  = (const float*)d_in[16];
    const float* W1    = (const float*)d_in[17];
    const float* b1    = (const float*)d_in[18];
    const float* W2    = (const float*)d_in[19];
    const float* b2    = (const float*)d_in[20];
    const float* W3    = (const float*)d_in[21];
    const float* b3    = (const float*)d_in[22];
    const float* ln2g  = (const float*)d_in[23];
    const float* ln2b  = (const float*)d_in[24];
    float* out = (float*)d_out;

    // ---- workspace carve-up (floats) ----
    float* ws = (float*)d_ws;
    size_t off = 0;
    float* qb   = ws + off; off += (size_t)NN * QO;
    float* kb   = ws + off; off += (size_t)NN * QO;
    float* vb   = ws + off; off += (size_t)NN * QO;
    float* qpb  = ws + off; off += (size_t)NN * PO;
    float* kpb  = ws + off; off += (size_t)NN * PO;
    float* vpb  = ws + off; off += (size_t)NN * PO;
    float* lg   = ws + off; off += (size_t)NE * HH;
    float* mbuf = ws + off; off += (size_t)NN * HH;
    float* sbuf = ws + off; off += (size_t)NN * HH;
    float* aggr = ws + off; off += (size_t)NN * PO;
    float* fall = ws + off; off += (size_t)NN * FIN;
    float* hpre = ws + off; off += (size_t)NN * DD;
    float* hbuf = ws + off; off += (size_t)NN * DD;
    float* m1   = ws + off; off += (size_t)NN * DD;
    float* m2   = ws + off; off += (size_t)NN * DD;
    float* m3   = ws + off; off += (size_t)NN * DD;

    // ---- init accumulators (fresh every call; harness does not re-poison) ----
    k_fill<<<cdiv(NN * FIN, 256), 256, 0, stream>>>(fall, 0.0f, NN * FIN);
    k_fill<<<cdiv(NN * PO, 256),  256, 0, stream>>>(aggr, 0.0f, NN * PO);
    k_fill<<<cdiv(NN * HH, 256),  256, 0, stream>>>(sbuf, 0.0f, NN * HH);
    k_fill<<<cdiv(NN * HH, 256),  256, 0, stream>>>(mbuf, -3.402823466e38f, NN * HH);

    // ---- projection GEMMs (WMMA, 16x32 tile per wave) ----
    auto gemm = [&](const float* A, const float* B, float* C, const float* bias,
                    int M, int K, int Nc, int epi) {
        int tiles  = (M >> 4) * (Nc >> 5);
        int blocks = cdiv(tiles, 8);
        k_gemm_wmma<<<blocks, 256, 0, stream>>>(A, B, C, bias, M, K, Nc, epi);
    };
    gemm(x, Wq,  qb,  nullptr, NN, DD, QO, 0);
    gemm(x, Wk,  kb,  nullptr, NN, DD, QO, 0);
    gemm(x, Wv,  vb,  nullptr, NN, DD, QO, 0);
    gemm(x, Wqp, qpb, nullptr, NN, DD, PO, 0);
    gemm(x, Wkp, kpb, nullptr, NN, DD, PO, 0);
    gemm(x, Wvp, vpb, nullptr, NN, DD, PO, 0);

    // ---- local -> global frames for point projections ----
    int npts = NN * 96;
    k_l2g<<<cdiv(npts, 256), 256, 0, stream>>>(qpb, R, t, npts);
    k_l2g<<<cdiv(npts, 256), 256, 0, stream>>>(kpb, R, t, npts);
    k_l2g<<<cdiv(npts, 256), 256, 0, stream>>>(vpb, R, t, npts);

    // ---- edge logits + scatter softmax (fused into aggregation) ----
    int eh = NE * HH;
    k_logits<<<cdiv(eh, 256), 256, 0, stream>>>(qb, kb, qpb, kpb, z, Wpair, sc, ei, lg, mbuf);
    k_aggregate<<<cdiv(eh, 256), 256, 0, stream>>>(lg, mbuf, sbuf, z, vb, vpb, ei, fall, aggr);
    k_alpha_norm<<<cdiv(NN * 768, 256), 256, 0, stream>>>(fall, sbuf);
    k_g2l_finish<<<cdiv(npts, 256), 256, 0, stream>>>(aggr, sbuf, R, t, fall);

    // ---- output projection + residual LN ----
    gemm(fall, Wo, hpre, bo, NN, FIN, DD, 1);
    k_ln<<<NN, DD, 0, stream>>>(x, hpre, ln1g, ln1b, hbuf);

    // ---- MLP + final LN ----
    gemm(hbuf, W1, m1, b1, NN, DD, DD, 2);   // relu
    gemm(m1,   W2, m2, b2, NN, DD, DD, 2);   // relu
    gemm(m2,   W3, m3, b3, NN, DD, DD, 1);
    k_ln<<<NN, DD, 0, stream>>>(hbuf, m3, ln2g, ln2b, out);
}